// FactorizedSpectralConv2d_53523882443339
// MI455X (gfx1250) — compile-verified
//
#include <hip/hip_runtime.h>
#include <hip/hip_bf16.h>
#include <math.h>

typedef float v2f __attribute__((ext_vector_type(2)));
typedef float v4f __attribute__((ext_vector_type(4)));
typedef float v8f __attribute__((ext_vector_type(8)));

#define PI_F 3.14159265358979323846f

// ---------------- problem constants ----------------
constexpr int B_    = 4;
constexpr int CIN   = 128;
constexpr int COUT  = 128;
constexpr int R2_   = 32;
constexpr int R3_   = 32;
constexpr int BCn   = B_ * CIN;   // 512
constexpr int BOn   = B_ * COUT;  // 512

// ---------------- workspace layout (floats) ----------------
constexpr size_t OFF_F1 = 0;                                   // 128 x 256
constexpr size_t OFF_G2 = OFF_F1 + (size_t)128 * 256;          // 256 x 512
constexpr size_t OFF_G4 = OFF_G2 + (size_t)256 * 512;          // 512 x 256
constexpr size_t OFF_G5 = OFF_G4 + (size_t)512 * 256;          // 256 x 128
constexpr size_t OFF_BT = OFF_G5 + (size_t)256 * 128;          // 2 x 4096 x 256
constexpr size_t OFF_AH = OFF_BT + (size_t)2 * 4096 * 256;     // 2 x 256 x 128
constexpr size_t OFF_MT = OFF_AH + (size_t)2 * 256 * 128;      // 2 x 2 x 32 x 4096
constexpr size_t OFF_XW = OFF_MT + (size_t)2 * 2 * 32 * 4096;  // BC x 2 x 256 x 64 (reused as Z)
constexpr size_t OFF_YF = OFF_XW + (size_t)BCn * 2 * 256 * 64; // BC x 2 x 128 x 64
constexpr size_t OFF_T  = OFF_YF + (size_t)BCn * 2 * 128 * 64; // 4096 x 2048
constexpr size_t OFF_U  = OFF_T  + (size_t)4096 * 2048;        // 128 x 2048
constexpr size_t OFF_OF = OFF_U  + (size_t)128 * 2048;         // BO x 2 x 128 x 64

// ---------------- table generation ----------------
__global__ __launch_bounds__(256)
void gen_dft_tables(float* __restrict__ ws) {
  int idx = blockIdx.x * 256 + threadIdx.x;
  const float c0 = 2.0f * PI_F / 256.0f;
  const float inv = 1.0f / 256.0f;
  if (idx < 128 * 256) {
    int m = idx >> 8, k = idx & 255;
    float t = c0 * (float)(((m & 63) * k) & 255);
    ws[OFF_F1 + idx] = (m < 64) ? __cosf(t) : -__sinf(t);
    return;
  }
  idx -= 128 * 256;
  if (idx < 256 * 512) {
    int row = idx >> 9, k = idx & 511;
    int part = row >> 7, hp = row & 127;
    int f = (hp < 64) ? hp : hp + 128;
    int kp = k >> 8, h = k & 255;
    float t = c0 * (float)((f * h) & 255);
    float v;
    if (part == 0) v = (kp == 0) ? __cosf(t) : __sinf(t);
    else           v = (kp == 0) ? -__sinf(t) : __cosf(t);
    ws[OFF_G2 + idx] = v;
    return;
  }
  idx -= 256 * 512;
  if (idx < 512 * 256) {
    int row = idx >> 8, k = idx & 255;
    int part = row >> 8, h = row & 255;
    int kp = k >> 7, m = k & 127;
    int f = (m < 64) ? m : m + 128;
    float t = c0 * (float)((h * f) & 255);
    float v;
    if (part == 0) v = (kp == 0) ? __cosf(t) : -__sinf(t);
    else           v = (kp == 0) ? __sinf(t) : __cosf(t);
    ws[OFF_G4 + idx] = v * inv;
    return;
  }
  idx -= 512 * 256;
  if (idx < 256 * 128) {
    int w = idx >> 7, k = idx & 127;
    float v;
    if (k == 0)       v = inv;
    else if (k < 64)  v = 2.0f * inv * __cosf(c0 * (float)((w * k) & 255));
    else if (k == 64) v = 0.0f;
    else              v = -2.0f * inv * __sinf(c0 * (float)((w * (k - 64)) & 255));
    ws[OFF_G5 + idx] = v;
    return;
  }
}

__global__ __launch_bounds__(256)
void gen_tt_tables(float* __restrict__ ws,
                   const float* __restrict__ a1, const float* __restrict__ b1,
                   const float* __restrict__ c1, const float* __restrict__ d1,
                   const float* __restrict__ a2, const float* __restrict__ b2,
                   const float* __restrict__ c2, const float* __restrict__ d2) {
  int idx = blockIdx.x * 256 + threadIdx.x;
  const int NBT = 2 * 4096 * 256;
  const int NAH = 2 * 256 * 128;
  const int NMT = 2 * 32 * 4096;
  if (idx < NBT) {
    int corner = idx / (4096 * 256);
    int rem = idx - corner * (4096 * 256);
    int row = rem >> 8, ccol = rem & 255;
    int part = row >> 11, r = row & 2047;
    int j = r >> 6, i = r & 63;
    int cpart = ccol >> 7, c = ccol & 127;
    const float* bf = corner ? b2 : b1;   // (64, 128, 32, 2)
    const float* e = bf + ((size_t)((i * CIN + c) * R2_ + j)) * 2;
    float re = e[0], im = e[1];
    ws[OFF_BT + idx] = (part == 0) ? ((cpart == 0) ? re : -im)
                                   : ((cpart == 0) ? im : re);
    return;
  }
  idx -= NBT;
  if (idx < NAH) {
    int corner = idx / (256 * 128);
    int rem = idx - corner * (256 * 128);
    int row = rem >> 7, icol = rem & 127;
    int part = row >> 7, o = row & 127;
    int ipart = icol >> 6, i = icol & 63;
    const float* af = corner ? a2 : a1;   // (1, 128, 64, 2)
    const float* e = af + ((size_t)(o * 64 + i)) * 2;
    float re = e[0], im = e[1];
    ws[OFF_AH + idx] = (part == 0) ? ((ipart == 0) ? re : -im)
                                   : ((ipart == 0) ? im : re);
    return;
  }
  idx -= NAH;
  if (idx < NMT) {
    int corner = idx / (32 * 4096);
    int rem = idx - corner * (32 * 4096);
    int j = rem >> 12, mode = rem & 4095;
    int x = mode >> 6, y = mode & 63;
    const float* cf = corner ? c2 : c1;   // (32, 64, 32, 2)
    const float* df = corner ? d2 : d1;   // (32, 64, 1, 2)
    float mre = 0.0f, mim = 0.0f;
    for (int k = 0; k < R3_; ++k) {
      const float* ce = cf + ((size_t)((j * 64 + x) * R3_ + k)) * 2;
      const float* de = df + ((size_t)(k * 64 + y)) * 2;
      float cr = ce[0], ci = ce[1], dr = de[0], di = de[1];
      mre = fmaf(cr, dr, mre); mre = fmaf(-ci, di, mre);
      mim = fmaf(cr, di, mim); mim = fmaf(ci, dr, mim);
    }
    size_t base = OFF_MT + (size_t)corner * 2 * 32 * 4096;
    ws[base + (size_t)j * 4096 + mode] = mre;
    ws[base + (size_t)32 * 4096 + (size_t)j * 4096 + mode] = mim;
    return;
  }
}

// ---------------- WMMA / async helpers ----------------
__device__ __forceinline__ v8f wmma4(v2f a, v2f b, v8f c) {
  return __builtin_amdgcn_wmma_f32_16x16x4_f32(
      false, a, false, b, (short)0, c, false, false);
}

__device__ __forceinline__ void async_b128(unsigned ldsaddr, const float* g) {
  asm volatile("global_load_async_to_lds_b128 %0, %1, off"
               :: "v"(ldsaddr), "v"((unsigned long long)(uintptr_t)g)
               : "memory");
}
__device__ __forceinline__ void wait_async0() {
  asm volatile("s_wait_asynccnt 0" ::: "memory");
}

// Stage codes: 0 W-DFT, 1 H-DFT, 2 GEMM1, 3 GEMM2, 4 inverse-H, 5 inverse-W.
// B addr(col, k=seg*SEGK+t) = colbase(col) + seg*SEGSTR + t*TSTR
template <int S> struct BTr;
template <> struct BTr<0> { static constexpr int SEGK = 256; static constexpr size_t TSTR = 1;     static constexpr size_t SEGSTR = 0; };
template <> struct BTr<1> { static constexpr int SEGK = 256; static constexpr size_t TSTR = 64;    static constexpr size_t SEGSTR = 16384; };
template <> struct BTr<2> { static constexpr int SEGK = 128; static constexpr size_t TSTR = 16384; static constexpr size_t SEGSTR = 8192; };
template <> struct BTr<3> { static constexpr int SEGK = 128; static constexpr size_t TSTR = 2048;  static constexpr size_t SEGSTR = 0; };
template <> struct BTr<4> { static constexpr int SEGK = 128; static constexpr size_t TSTR = 64;    static constexpr size_t SEGSTR = 8192; };
template <> struct BTr<5> { static constexpr int SEGK = 64;  static constexpr size_t TSTR = 1;     static constexpr size_t SEGSTR = 16384; };

template <int S>
__device__ __forceinline__ size_t colbase(int gcol, int lcol, int corner) {
  if constexpr (S == 0) {
    return (size_t)gcol * 256;
  } else if constexpr (S == 1) {
    int bc = gcol >> 6, w = gcol & 63;
    return (size_t)bc * 32768 + w;
  } else if constexpr (S == 2) {
    int b = gcol >> 12, mode = gcol & 4095;
    int hp = corner * 64 + (mode >> 6);
    return (size_t)b * 2097152 + (size_t)hp * 64 + (mode & 63);
  } else if constexpr (S == 3) {
    return (size_t)lcol;
  } else if constexpr (S == 4) {
    int bo = gcol >> 6, w = gcol & 63;
    return (size_t)bo * 16384 + w;
  } else {
    int bo = gcol >> 8, h = gcol & 255;
    return (size_t)bo * 32768 + (size_t)h * 64;
  }
}

template <int S>
__device__ __forceinline__ void cst(float* __restrict__ Cdst,
                                    int m, int gcol, int lcol, int corner, float v) {
  if constexpr (S == 0) {
    int part = m >> 6, mm = m & 63;
    int bc = gcol >> 8, h = gcol & 255;
    Cdst[((size_t)(bc * 2 + part)) * 16384 + (size_t)h * 64 + mm] = v;
  } else if constexpr (S == 1) {
    int part = m >> 7, hp = m & 127;
    int bc = gcol >> 6, w = gcol & 63;
    Cdst[((size_t)(bc * 2 + part)) * 8192 + (size_t)hp * 64 + w] = v;
  } else if constexpr (S == 2) {
    Cdst[(size_t)m * 2048 + lcol] = v;
  } else if constexpr (S == 3) {
    int part = m >> 7, o = m & 127;
    int b = gcol >> 12, mode = gcol & 4095;
    int hp = corner * 64 + (mode >> 6);
    Cdst[((size_t)((b * COUT + o) * 2 + part)) * 8192 + (size_t)hp * 64 + (mode & 63)] = v;
  } else if constexpr (S == 4) {
    int part = m >> 8, h = m & 255;
    int bo = gcol >> 6, w = gcol & 63;
    Cdst[((size_t)(bo * 2 + part)) * 16384 + (size_t)h * 64 + w] = v;
  } else {
    int bo = gcol >> 8, h = gcol & 255;
    Cdst[(size_t)bo * 65536 + (size_t)h * 256 + m] = v;
  }
}

// Async-to-LDS double-buffered WMMA GEMM.
// Workgroup tile 128(M) x 64(N), K chunk 32.  8 waves = 4(M) x 2(N); each wave
// owns a 32x32 C block (2x2 WMMA tiles).  Global tiles stream directly into
// LDS via GLOBAL_LOAD_ASYNC_TO_LDS_B128 (ASYNCcnt), compute reads ds_load.
// A (and k-contig B) layout [row][36]: 16B-aligned async writes, conflict-free
// b64 frag reads.  Col-contig B stays untransposed as [k][72]: b32 frag reads,
// hi-lane rows land +16 banks -> conflict-free.
template <int S>
__global__ __launch_bounds__(256) __attribute__((amdgpu_waves_per_eu(4)))
void gemm_wmma(const float* __restrict__ A, const float* __restrict__ Bsrc,
               float* __restrict__ Cdst, int K, int col0, int corner) {
  constexpr int MT = 128, NT = 64, KC = 32;
  constexpr int LDA = 36;                 // [row][36] stride
  constexpr int LDB = 72;                 // [k][72] stride (col-contig B)
  constexpr size_t TSTR = BTr<S>::TSTR, SEGSTR = BTr<S>::SEGSTR;
  constexpr int SEGK = BTr<S>::SEGK;
  constexpr bool KCONTIG = (TSTR == 1);

  __shared__ float As[2][MT * LDA];       // 2 x 4608 floats
  __shared__ float Bs[2][2304];           // 64*36 or 32*72

  const int tid = threadIdx.x;
  const int lane = tid & 31, wave = tid >> 5;
  const int wm = wave & 3, wn = wave >> 2;
  const int mtile = blockIdx.y * MT;
  const int ncol0 = blockIdx.x * NT;
  const int hi = lane >> 4, hioff = hi * 2;

  const int lcol0 = ncol0 + wn * 32 + (lane & 15);
  const int lcol1 = lcol0 + 16;
  const int gcol0 = col0 + lcol0, gcol1 = col0 + lcol1;

  // per-thread async staging coordinates
  int am[4], ak[4];
#pragma unroll
  for (int i = 0; i < 4; ++i) {
    int v = tid + i * 256;                // MT*KC/4 = 1024 b128 transfers
    am[i] = v >> 3;                       // row 0..127
    ak[i] = (v & 7) * 4;                  // k4
  }
  int bq0[2], bq1[2];
  size_t bcb[2];
#pragma unroll
  for (int i = 0; i < 2; ++i) {
    int v = tid + i * 256;                // NT*KC/4 = 512 b128 transfers
    if constexpr (KCONTIG) {
      bq0[i] = v >> 3;                    // col 0..63
      bq1[i] = (v & 7) * 4;               // k4
      int c = ncol0 + bq0[i];
      bcb[i] = colbase<S>(col0 + c, c, corner);
    } else {
      bq0[i] = v >> 4;                    // k row 0..31
      bq1[i] = (v & 15) * 4;              // col4
      bcb[i] = 0;
    }
  }
  const size_t bbase = colbase<S>(col0 + ncol0, ncol0, corner);

  auto issueAsync = [&](int buf, int k0) {
    const int seg = k0 / SEGK;
    const int toff = k0 % SEGK;
#pragma unroll
    for (int i = 0; i < 4; ++i) {
      unsigned l = (unsigned)(uintptr_t)&As[buf][am[i] * LDA + ak[i]];
      async_b128(l, A + (size_t)(mtile + am[i]) * K + k0 + ak[i]);
    }
#pragma unroll
    for (int i = 0; i < 2; ++i) {
      if constexpr (KCONTIG) {
        unsigned l = (unsigned)(uintptr_t)&Bs[buf][bq0[i] * LDA + bq1[i]];
        async_b128(l, Bsrc + bcb[i] + (size_t)seg * SEGSTR + toff + bq1[i]);
      } else {
        unsigned l = (unsigned)(uintptr_t)&Bs[buf][bq0[i] * LDB + bq1[i]];
        async_b128(l, Bsrc + bbase + (size_t)seg * SEGSTR +
                      (size_t)(toff + bq0[i]) * TSTR + bq1[i]);
      }
    }
  };

  v8f acc00 = {}, acc01 = {}, acc10 = {}, acc11 = {};

  const int nq = K / KC;
  issueAsync(0, 0);
  wait_async0();
  __syncthreads();
  for (int q = 0; q < nq; ++q) {
    const int buf = q & 1;
    if (q + 1 < nq) issueAsync(buf ^ 1, (q + 1) * KC);  // overlaps compute

    const float* pa0 = &As[buf][(wm * 32 + (lane & 15)) * LDA + hioff];
    const float* pa1 = pa0 + 16 * LDA;
    const float* pb0;
    const float* pb1;
    if constexpr (KCONTIG) {
      pb0 = &Bs[buf][(wn * 32 + (lane & 15)) * LDA + hioff];
      pb1 = pb0 + 16 * LDA;
    } else {
      pb0 = &Bs[buf][hioff * LDB + wn * 32 + (lane & 15)];
      pb1 = pb0 + 16;
    }
#pragma unroll
    for (int t = 0; t < KC; t += 4) {
      v2f a0 = *(const v2f*)(pa0 + t);
      v2f a1 = *(const v2f*)(pa1 + t);
      v2f b0, b1;
      if constexpr (KCONTIG) {
        b0 = *(const v2f*)(pb0 + t);
        b1 = *(const v2f*)(pb1 + t);
      } else {
        b0.x = pb0[(size_t)t * LDB];
        b0.y = pb0[(size_t)(t + 1) * LDB];
        b1.x = pb1[(size_t)t * LDB];
        b1.y = pb1[(size_t)(t + 1) * LDB];
      }
      acc00 = wmma4(a0, b0, acc00);
      acc10 = wmma4(a1, b0, acc10);
      acc01 = wmma4(a0, b1, acc01);
      acc11 = wmma4(a1, b1, acc11);
    }
    if (q + 1 < nq) {
      wait_async0();                      // this wave's async writes landed
      __syncthreads();                    // visible to all waves
    }
  }

#pragma unroll
  for (int r = 0; r < 8; ++r) {
    int m0 = mtile + hi * 8 + r;
    int m1 = m0 + 16;
    cst<S>(Cdst, m0, gcol0, lcol0, corner, acc00[r]);
    cst<S>(Cdst, m1, gcol0, lcol0, corner, acc10[r]);
    cst<S>(Cdst, m0, gcol1, lcol1, corner, acc01[r]);
    cst<S>(Cdst, m1, gcol1, lcol1, corner, acc11[r]);
  }
}

// ---------------- j-reduction ----------------
__global__ __launch_bounds__(256)
void ureduce_kernel(const float* __restrict__ T, const float* __restrict__ Mtab,
                    float* __restrict__ U, int col0, int corner) {
  int idx = blockIdx.x * 256 + threadIdx.x;   // 64 * 2048
  int i = idx >> 11;
  int lcol = idx & 2047;
  int mode = (col0 + lcol) & 4095;
  const float* Mre = Mtab + (size_t)corner * 2 * 32 * 4096;
  const float* Mim = Mre + (size_t)32 * 4096;
  float ure = 0.0f, uim = 0.0f;
#pragma unroll 4
  for (int j = 0; j < R2_; ++j) {
    float tre = T[((size_t)(j * 64 + i)) * 2048 + lcol];
    float tim = T[((size_t)(2048 + j * 64 + i)) * 2048 + lcol];
    float mre = Mre[(size_t)j * 4096 + mode];
    float mim = Mim[(size_t)j * 4096 + mode];
    ure = fmaf(mre, tre, ure); ure = fmaf(-mim, tim, ure);
    uim = fmaf(mre, tim, uim); uim = fmaf(mim, tre, uim);
  }
  U[(size_t)i * 2048 + lcol] = ure;
  U[((size_t)(64 + i)) * 2048 + lcol] = uim;
}

// ---------------- launch ----------------
extern "C" void kernel_launch(void* const* d_in, const int* in_sizes, int n_in,
                              void* d_out, int out_size, void* d_ws, size_t ws_size,
                              hipStream_t stream) {
  const float* x  = (const float*)d_in[0];
  const float* a1 = (const float*)d_in[1];
  const float* b1 = (const float*)d_in[2];
  const float* c1 = (const float*)d_in[3];
  const float* d1 = (const float*)d_in[4];
  const float* a2 = (const float*)d_in[5];
  const float* b2 = (const float*)d_in[6];
  const float* c2 = (const float*)d_in[7];
  const float* d2 = (const float*)d_in[8];
  float* ws  = (float*)d_ws;
  float* out = (float*)d_out;

  gen_dft_tables<<<dim3((128*256 + 256*512 + 512*256 + 256*128 + 255) / 256), 256, 0, stream>>>(ws);
  {
    int ntot = 2 * 4096 * 256 + 2 * 256 * 128 + 2 * 32 * 4096;
    gen_tt_tables<<<dim3((ntot + 255) / 256), 256, 0, stream>>>(ws, a1, b1, c1, d1, a2, b2, c2, d2);
  }

  // W-DFT: C[128 x 131072]
  gemm_wmma<0><<<dim3(131072 / 64, 1), 256, 0, stream>>>(ws + OFF_F1, x, ws + OFF_XW, 256, 0, 0);

  // H-DFT: C[256 x 32768]
  gemm_wmma<1><<<dim3(32768 / 64, 2), 256, 0, stream>>>(ws + OFF_G2, ws + OFF_XW, ws + OFF_YF, 512, 0, 0);

  // TT contraction per corner, N chunked
  for (int corner = 0; corner < 2; ++corner) {
    const float* BT = ws + OFF_BT + (size_t)corner * 4096 * 256;
    const float* AH = ws + OFF_AH + (size_t)corner * 256 * 128;
    for (int chunk = 0; chunk < 8; ++chunk) {
      int col0 = chunk * 2048;
      gemm_wmma<2><<<dim3(2048 / 64, 4096 / 128), 256, 0, stream>>>(BT, ws + OFF_YF, ws + OFF_T, 256, col0, corner);
      ureduce_kernel<<<dim3((64 * 2048) / 256), 256, 0, stream>>>(ws + OFF_T, ws + OFF_MT, ws + OFF_U, col0, corner);
      gemm_wmma<3><<<dim3(2048 / 64, 256 / 128), 256, 0, stream>>>(AH, ws + OFF_U, ws + OFF_OF, 128, col0, corner);
    }
  }

  // inverse H: C[512 x 32768] into Z (= XW, now dead)
  gemm_wmma<4><<<dim3(32768 / 64, 512 / 128), 256, 0, stream>>>(ws + OFF_G4, ws + OFF_OF, ws + OFF_XW, 256, 0, 0);

  // inverse W: C[256 x 131072] straight into d_out
  gemm_wmma<5><<<dim3(131072 / 64, 256 / 128), 256, 0, stream>>>(ws + OFF_G5, ws + OFF_XW, out, 128, 0, 0);
}